// SSP_16836271800917
// MI455X (gfx1250) — compile-verified
//
#include <hip/hip_runtime.h>
#include <stdint.h>

// SPP max-pool (k=5,9,13, stride-1 SAME) + concat, NHWC f32.
// Memory-bound: 64 MiB in (L2-resident on MI455X's 192MB L2), 256 MiB out
// -> ~13.7us HBM floor at 23.3 TB/s; this kernel streams input once per block
// via the CDNA5 async global->LDS path and writes output once with NT stores.
//
// CDNA5 paths: global_load_async_to_lds_b64 (ASYNCcnt) with software-pipelined
// prefetch (partial s_wait_asynccnt), wave32 layout, VOPD-friendly f32x2 math.

typedef __attribute__((ext_vector_type(2))) float f32x2;

#define N_      16
#define H_      64
#define W_      64
#define C_      256
#define OC_     1024
#define CCHUNK  64            // channels per block
#define NC2     32            // float2 groups per block (CCHUNK/2)
#define WT      16            // output w positions per block
#define WS      (WT + 12)     // staged w positions (halo 6 each side)
#define EL      (WS * NC2)    // staged float2 elements per row = 896
#define NBUF    3             // triple buffer: prefetch writer vs lagging reader
#define THREADS 512

static __device__ __forceinline__ f32x2 vmax2(f32x2 a, f32x2 b) {
  return __builtin_elementwise_max(a, b);
}

// Async copy 8 bytes global -> LDS, per lane. Tracked by ASYNCcnt.
static __device__ __forceinline__ void async_ld_b64(uint32_t lds_addr, const void* gptr) {
  asm volatile("global_load_async_to_lds_b64 %0, %1, off"
               :: "v"(lds_addr), "v"(gptr)
               : "memory");
}
// Wait until <= N async ops outstanding (async ops retire in order).
static __device__ __forceinline__ void wait_async_le1() {
  asm volatile("s_wait_asynccnt 0x1" ::: "memory");
}
static __device__ __forceinline__ void wait_async_le2() {
  asm volatile("s_wait_asynccnt 0x2" ::: "memory");
}

__global__ __launch_bounds__(THREADS)
void spp_kernel(const float* __restrict__ x, float* __restrict__ out) {
  __shared__ f32x2 buf[NBUF][EL];   // 3 x 7168 B = 21504 B

  const int tid = threadIdx.x;
  const int n   = blockIdx.x;
  const int c0  = blockIdx.y * CCHUNK;
  const int w0  = blockIdx.z * WT;

  const int c2 = tid & (NC2 - 1);   // float2 channel group (lanes fastest -> coalesced)
  const int wl = tid >> 5;          // 0..15 output w within tile
  const int w  = w0 + wl;
  const int cl = c2 * 2;

  // Per-lane staging source offsets (float offsets within one input row).
  // Horizontal SAME padding via index clamp (equivalent to -inf pad for max).
  const int sl0 = tid >> 5, cp0 = tid & 31;
  const int gw0 = min(max(w0 - 6 + sl0, 0), W_ - 1);
  const int goff0 = gw0 * C_ + c0 + 2 * cp0;
  const int e1 = tid + THREADS;
  const int sl1 = e1 >> 5, cp1 = e1 & 31;
  const int gw1 = min(max(w0 - 6 + sl1, 0), W_ - 1);
  const int goff1 = gw1 * C_ + c0 + 2 * cp1;
  const bool do1 = (e1 < EL);       // wave-uniform (384 = 12 waves)

  const uint32_t lds_base = (uint32_t)(uintptr_t)(&buf[0][0]);
  const uint32_t la0 = (uint32_t)tid * 8u;
  const uint32_t la1 = (uint32_t)e1 * 8u;
  const size_t   colbase = (size_t)w * OC_ + (size_t)(c0 + cl);
  const size_t   rowstr  = (size_t)W_ * OC_;
  const size_t   xrowstr = (size_t)W_ * C_;
  const float*   xn      = x + (size_t)n * H_ * xrowstr;

  // Vertical sliding windows (register ring buffers).
  f32x2 r5[5] = {}, r9[9] = {}, r13[13] = {};

  // ---- prologue: stage row for s=0 (virtual row -6 -> clamped 0) into buf[0]
  {
    const float* rowp = xn;   // r = 0
    async_ld_b64(lds_base + la0, rowp + goff0);
    if (do1) async_ld_b64(lds_base + la1, rowp + goff1);
  }

  int br = 0;                        // buffer holding row s
  for (int s = 0; s < H_ + 12; ++s) {
    // ---- prefetch row s+1 into next buffer (overlaps with compute of row s)
    int bw = br + 1; if (bw == NBUF) bw = 0;
    {
      const int r1 = min(max(s + 1 - 6, 0), H_ - 1);
      const float* rowp1 = xn + (size_t)r1 * xrowstr;
      const uint32_t lbw = lds_base + (uint32_t)bw * (EL * 8);
      async_ld_b64(lbw + la0, rowp1 + goff0);
      if (do1) {
        async_ld_b64(lbw + la1, rowp1 + goff1);
        wait_async_le2();            // row-s ops retired; s+1 still in flight
      } else {
        wait_async_le1();
      }
    }
    __syncthreads();                 // row s visible to all waves

    // ---- horizontal pass: h5 -> h9 -> h13 (incremental) ----
    const f32x2* row2 = &buf[br][0];
    f32x2 v[13];
#pragma unroll
    for (int j = 0; j < 13; ++j) v[j] = row2[(wl + j) * NC2 + c2];

    const f32x2 h1 = vmax2(vmax2(vmax2(v[4], v[5]), vmax2(v[6], v[7])), v[8]);
    const f32x2 h2 = vmax2(h1, vmax2(vmax2(v[2], v[3]), vmax2(v[9], v[10])));
    const f32x2 h3 = vmax2(h2, vmax2(vmax2(v[0], v[1]), vmax2(v[11], v[12])));

    // ---- push into vertical rings ----
#pragma unroll
    for (int i = 0; i < 4; ++i) r5[i] = r5[i + 1];
    r5[4] = h1;
#pragma unroll
    for (int i = 0; i < 8; ++i) r9[i] = r9[i + 1];
    r9[8] = h2;
#pragma unroll
    for (int i = 0; i < 12; ++i) r13[i] = r13[i + 1];
    r13[12] = h3;

    // ---- emit (each pool at its own row phase; NT streamed stores) ----
    if (s >= 6 && s < 6 + H_) {   // x copy, h = s-6, channels [768,1024)
      const size_t idx = (size_t)(n * H_ + (s - 6)) * rowstr + colbase + 768;
      __builtin_nontemporal_store(v[6], (f32x2*)(out + idx));
    }
    if (s >= 8 && s < 8 + H_) {   // p5, h = s-8, channels [512,768)
      const f32x2 p5 = vmax2(vmax2(vmax2(r5[0], r5[1]), vmax2(r5[2], r5[3])), r5[4]);
      const size_t idx = (size_t)(n * H_ + (s - 8)) * rowstr + colbase + 512;
      __builtin_nontemporal_store(p5, (f32x2*)(out + idx));
    }
    if (s >= 10 && s < 10 + H_) { // p9, h = s-10, channels [256,512)
      const f32x2 p9 = vmax2(vmax2(vmax2(vmax2(r9[0], r9[1]), vmax2(r9[2], r9[3])),
                                   vmax2(vmax2(r9[4], r9[5]), vmax2(r9[6], r9[7]))),
                             r9[8]);
      const size_t idx = (size_t)(n * H_ + (s - 10)) * rowstr + colbase + 256;
      __builtin_nontemporal_store(p9, (f32x2*)(out + idx));
    }
    if (s >= 12) {                // p13, h = s-12, channels [0,256)
      const f32x2 a = vmax2(vmax2(vmax2(r13[0], r13[1]), vmax2(r13[2], r13[3])),
                            vmax2(vmax2(r13[4], r13[5]), vmax2(r13[6], r13[7])));
      const f32x2 b = vmax2(vmax2(vmax2(r13[8], r13[9]), vmax2(r13[10], r13[11])),
                            r13[12]);
      const f32x2 p13 = vmax2(a, b);
      const size_t idx = (size_t)(n * H_ + (s - 12)) * rowstr + colbase;
      __builtin_nontemporal_store(p13, (f32x2*)(out + idx));
    }

    br = bw;
  }
}

extern "C" void kernel_launch(void* const* d_in, const int* in_sizes, int n_in,
                              void* d_out, int out_size, void* d_ws, size_t ws_size,
                              hipStream_t stream) {
  (void)in_sizes; (void)n_in; (void)out_size; (void)d_ws; (void)ws_size;
  const float* x = (const float*)d_in[0];
  float* out = (float*)d_out;
  dim3 grid(N_, C_ / CCHUNK, W_ / WT);   // (16, 4, 4) = 256 blocks x 16 wave32
  spp_kernel<<<grid, THREADS, 0, stream>>>(x, out);
}